// GCF_65910568124533
// MI455X (gfx1250) — compile-verified
//
#include <hip/hip_runtime.h>
#include <hip/hip_bf16.h>

typedef __attribute__((ext_vector_type(2))) float v2f;
typedef __attribute__((ext_vector_type(8))) float v8f;

#define DD     128
#define SLOPE  0.01f
#define EPSN   1e-12f

// ---------------------------------------------------------------------------
// Copy uEmbd / iEmbd into columns [0,128) of finalEmbd (row stride = fw).
// ---------------------------------------------------------------------------
__global__ void init_embed(const float* __restrict__ uE,
                           const float* __restrict__ iE,
                           float* __restrict__ fe,
                           int numUsers, int n, int fw) {
    size_t i = (size_t)blockIdx.x * blockDim.x + threadIdx.x;
    size_t total = (size_t)n * DD;
    if (i >= total) return;
    int row = (int)(i >> 7);
    int col = (int)(i & 127);
    float v = (row < numUsers) ? uE[(size_t)row * DD + col]
                               : iE[(size_t)(row - numUsers) * DD + col];
    fe[(size_t)row * fw + col] = v;
}

// ---------------------------------------------------------------------------
// SpMM over COO edges: Lx[rows[e], :] += vals[e] * F[cols[e], :]
// One wave (32 lanes) per edge; each lane handles 4 contiguous floats.
// F has row stride fw (a column slice of finalEmbd); Lx has row stride 128.
// ---------------------------------------------------------------------------
__global__ void spmm_coo(const int* __restrict__ rows,
                         const int* __restrict__ cols,
                         const float* __restrict__ vals,
                         const float* __restrict__ F,
                         float* __restrict__ Lx,
                         int nnz, int fw) {
    int e    = blockIdx.x * 8 + (threadIdx.x >> 5);
    int lane = threadIdx.x & 31;
    if (e >= nnz) return;
    int   r = rows[e];
    int   c = cols[e];
    float v = vals[e];
    int   d0 = lane * 4;
    const float4 f = *(const float4*)(F + (size_t)c * fw + d0);
    float* dst = Lx + (size_t)r * DD + d0;
    atomicAdd(dst + 0, v * f.x);
    atomicAdd(dst + 1, v * f.y);
    atomicAdd(dst + 2, v * f.z);
    atomicAdd(dst + 3, v * f.w);
}

// ---------------------------------------------------------------------------
// Fused layer: h = leakyrelu((Lx+f)@Wlin + blin + (Lx*f)@Wint + bint)
//              Fout = h / max(||h||_2, eps)     (row-wise)
// Block = 256 threads (8 waves). Block owns 16 complete rows (all 128 cols);
// wave w computes the 16x16 tile at columns [16w, 16w+16) with
// v_wmma_f32_16x16x4_f32 accumulating both GEMMs into one accumulator.
// ---------------------------------------------------------------------------
__global__ void gcf_layer(const float* __restrict__ Fin,   // stride fw
                          const float* __restrict__ Lx,    // stride 128
                          const float* __restrict__ Wlin,  // [128,128]
                          const float* __restrict__ blin,  // [128]
                          const float* __restrict__ Wint,  // [128,128]
                          const float* __restrict__ bint,  // [128]
                          float* __restrict__ Fout,        // stride fw
                          int nrows, int fw) {
    __shared__ float sA1[16 * DD];
    __shared__ float sA2[16 * DD];
    __shared__ float sOut[16 * DD];
    __shared__ float sScale[16];

    const int tid  = threadIdx.x;
    const int row0 = blockIdx.x * 16;

    // Stage A1 = Lx + f, A2 = Lx * f into LDS (16 x 128 each).
    for (int i = tid; i < 16 * DD; i += 256) {
        int row = i >> 7, col = i & 127;
        int g = row0 + row;
        float f  = 0.0f, lx = 0.0f;
        if (g < nrows) {
            f  = Fin[(size_t)g * fw + col];
            lx = Lx [(size_t)g * DD + col];
        }
        sA1[i] = lx + f;
        sA2[i] = lx * f;
    }
    __syncthreads();

    const int wave = tid >> 5;
    const int lane = tid & 31;
    const int n0   = wave * 16;            // column tile base
    const int m    = lane & 15;            // A row within tile
    const int koff = (lane >> 4) * 2;      // K offset: lanes 0-15 -> K0/K1, 16-31 -> K2/K3
    const int ncol = n0 + (lane & 15);     // global column for B / C / D

    v8f acc = {0.f, 0.f, 0.f, 0.f, 0.f, 0.f, 0.f, 0.f};

    for (int k4 = 0; k4 < DD / 4; ++k4) {
        const int kb = k4 * 4 + koff;

        v2f a1; a1.x = sA1[m * DD + kb];  a1.y = sA1[m * DD + kb + 1];
        v2f b1; b1.x = Wlin[(size_t)kb * DD + ncol];
                b1.y = Wlin[(size_t)(kb + 1) * DD + ncol];
        acc = __builtin_amdgcn_wmma_f32_16x16x4_f32(false, a1, false, b1,
                                                    (short)0, acc, false, false);

        v2f a2; a2.x = sA2[m * DD + kb];  a2.y = sA2[m * DD + kb + 1];
        v2f b2; b2.x = Wint[(size_t)kb * DD + ncol];
                b2.y = Wint[(size_t)(kb + 1) * DD + ncol];
        acc = __builtin_amdgcn_wmma_f32_16x16x4_f32(false, a2, false, b2,
                                                    (short)0, acc, false, false);
    }

    // Bias + LeakyReLU, then park the tile in LDS (C/D layout: VGPR v holds
    // row v (lanes 0-15) / row v+8 (lanes 16-31), column = lane&15).
    const float bias  = blin[ncol] + bint[ncol];
    const int   rbase = (lane >> 4) * 8;
#pragma unroll
    for (int v = 0; v < 8; ++v) {
        float x = acc[v] + bias;
        x = (x > 0.0f) ? x : SLOPE * x;
        sOut[(v + rbase) * DD + ncol] = x;
    }
    __syncthreads();

    // Row-wise L2 norm (block owns full rows).
    if (tid < 16) {
        float ss = 0.0f;
        for (int c = 0; c < DD; ++c) {
            float x = sOut[tid * DD + c];
            ss += x * x;
        }
        sScale[tid] = 1.0f / fmaxf(sqrtf(ss), EPSN);
    }
    __syncthreads();

    for (int i = tid; i < 16 * DD; i += 256) {
        int row = i >> 7, col = i & 127;
        int g = row0 + row;
        if (g < nrows)
            Fout[(size_t)g * fw + col] = sOut[i] * sScale[row];
    }
}

// ---------------------------------------------------------------------------
// out[b] = dot(finalEmbd[u[b], :], finalEmbd[i[b]+numUsers, :]) over fw cols.
// One wave per output element.
// ---------------------------------------------------------------------------
__global__ void dot_rows(const float* __restrict__ fe,
                         const int* __restrict__ uIdx,
                         const int* __restrict__ iIdx,
                         float* __restrict__ out,
                         int batch, int numUsers, int fw) {
    int b    = blockIdx.x * 8 + (threadIdx.x >> 5);
    int lane = threadIdx.x & 31;
    if (b >= batch) return;
    const float* pu = fe + (size_t)uIdx[b] * fw;
    const float* pi = fe + (size_t)(iIdx[b] + numUsers) * fw;
    float s = 0.0f;
    for (int j = lane; j < fw; j += 32)
        s += pu[j] * pi[j];
#pragma unroll
    for (int off = 16; off >= 1; off >>= 1)
        s += __shfl_xor(s, off, 32);
    if (lane == 0) out[b] = s;
}

// ---------------------------------------------------------------------------
extern "C" void kernel_launch(void* const* d_in, const int* in_sizes, int n_in,
                              void* d_out, int out_size, void* d_ws, size_t ws_size,
                              hipStream_t stream) {
    const int*   userIdx = (const int*)  d_in[0];
    const int*   itemIdx = (const int*)  d_in[1];
    const int*   rows    = (const int*)  d_in[2];
    const int*   cols    = (const int*)  d_in[3];
    const float* vals    = (const float*)d_in[4];
    const float* uEmbd   = (const float*)d_in[5];
    const float* iEmbd   = (const float*)d_in[6];
    const float* Wlin    = (const float*)d_in[7];
    const float* blin    = (const float*)d_in[8];
    const float* Wint    = (const float*)d_in[9];
    const float* bint    = (const float*)d_in[10];

    const int batch    = in_sizes[0];
    const int nnz      = in_sizes[2];
    const int numUsers = in_sizes[5] / DD;
    const int numItems = in_sizes[6] / DD;
    const int n        = numUsers + numItems;
    const int nLayers  = in_sizes[8] / DD;          // blin is [L, 128]
    const int fw       = DD * (nLayers + 1);        // finalEmbd width (512)

    float* finalEmbd = (float*)d_ws;                       // [n, fw]
    float* Lx        = finalEmbd + (size_t)n * fw;         // [n, 128]

    // 0) initial features into finalEmbd[:, 0:128]
    {
        size_t total = (size_t)n * DD;
        int blocks = (int)((total + 255) / 256);
        init_embed<<<blocks, 256, 0, stream>>>(uEmbd, iEmbd, finalEmbd,
                                               numUsers, n, fw);
    }

    // 1..L) layers
    for (int l = 0; l < nLayers; ++l) {
        const float* Fin  = finalEmbd + (size_t)l * DD;
        float*       Fout = finalEmbd + (size_t)(l + 1) * DD;

        hipMemsetAsync(Lx, 0, (size_t)n * DD * sizeof(float), stream);

        int spmmBlocks = (nnz + 7) / 8;
        spmm_coo<<<spmmBlocks, 256, 0, stream>>>(rows, cols, vals, Fin, Lx,
                                                 nnz, fw);

        int gemmBlocks = (n + 15) / 16;
        gcf_layer<<<gemmBlocks, 256, 0, stream>>>(
            Fin, Lx,
            Wlin + (size_t)l * DD * DD, blin + (size_t)l * DD,
            Wint + (size_t)l * DD * DD, bint + (size_t)l * DD,
            Fout, n, fw);
    }

    // final) batched dot products
    {
        int blocks = (batch + 7) / 8;
        dot_rows<<<blocks, 256, 0, stream>>>(finalEmbd, userIdx, itemIdx,
                                             (float*)d_out, batch, numUsers, fw);
    }
}